// AttentionSampling_3341484556842
// MI455X (gfx1250) — compile-verified
//
#include <hip/hip_runtime.h>
#include <math.h>

// ---------------------------------------------------------------------------
// AttentionSampling for MI455X (gfx1250, wave32, WMMA)
//   X = normalize(embeds);  S_sem = X @ X^T  (fp32 WMMA 16x16x4, exact)
//   S  = 1.5*S_sem[i,:] + 0.5*sum_{j: adj[i,j]!=0} S_sem[j,:]   (sparse gather)
//   diag = -inf; per-row top-32; COO output; sorted unique w/ sentinel N
// ---------------------------------------------------------------------------

#define N_NODES 4096
#define D_DIM   128
#define TOPK    32

typedef float v2f __attribute__((ext_vector_type(2)));
typedef float v8f __attribute__((ext_vector_type(8)));

// workspace layout (float units)
#define WS_X_OFF    0                                   // 4096*128  = 2 MB
#define WS_S_OFF    (N_NODES * D_DIM)                   // 4096*4096 = 64 MB
#define WS_FLAG_OFF (WS_S_OFF + (size_t)N_NODES * N_NODES)

// output layout (float units): sampled_nodes | indices[2,N*K] | values[N*K]
#define OUT_SAMP_OFF 0
#define OUT_IDX_OFF  (N_NODES)
#define OUT_VAL_OFF  (N_NODES + 2 * N_NODES * TOPK)

// --------------------------- row normalize ---------------------------------
__global__ void __launch_bounds__(128)
k_normalize(const float* __restrict__ embeds, float* __restrict__ X,
            int* __restrict__ flags) {
  __shared__ float red[128];
  const int row = blockIdx.x;
  const int t   = threadIdx.x;
  float v = embeds[row * D_DIM + t];
  red[t] = v * v;
  __syncthreads();
  #pragma unroll
  for (int off = 64; off > 0; off >>= 1) {
    if (t < off) red[t] += red[t + off];
    __syncthreads();
  }
  float nrm = fmaxf(sqrtf(red[0]), 1e-12f);
  X[row * D_DIM + t] = v / nrm;
  if (t == 0) flags[row] = 0;  // reset unique-flags for this call
}

// --------------------------- S_sem = X X^T via WMMA f32 --------------------
// grid: (32, 256); block 256 = 8 waves; wave w -> tileN = bx*8+w, tileM = by.
__global__ void __launch_bounds__(256)
k_gemm_xxt(const float* __restrict__ X, float* __restrict__ S) {
  __shared__ float ldsA[16 * 130];  // stride 130 -> conflict-free banks
  const int tileM = blockIdx.y;
  const int t     = threadIdx.x;

  // cooperative load of the 16x128 A tile
  for (int e = t; e < 16 * D_DIM; e += 256) {
    const int r = e >> 7, c = e & 127;
    ldsA[r * 130 + c] = X[((size_t)tileM * 16 + r) * D_DIM + c];
  }
  __syncthreads();

  const int wave  = t >> 5;
  const int lane  = t & 31;
  const int tileN = blockIdx.x * 8 + wave;
  const int half  = lane >> 4;   // 0: K pair {0,1}; 1: K pair {2,3}
  const int l16   = lane & 15;

  const float* Brow = X + ((size_t)tileN * 16 + l16) * D_DIM;

  v8f acc = {};
  #pragma unroll
  for (int k = 0; k < D_DIM; k += 4) {
    const int kk = k + 2 * half;                       // even -> 8B aligned
    v2f a = *(const v2f*)&ldsA[l16 * 130 + kk];        // A[M=l16][kk..kk+1]
    v2f b = *(const v2f*)&Brow[kk];                    // B[kk..kk+1][N=l16]
    acc = __builtin_amdgcn_wmma_f32_16x16x4_f32(
        false, a, false, b, (short)0, acc, false, false);
  }

  // C/D layout: VGPR r -> M = r + 8*half, N = l16
  const size_t nCol = (size_t)tileN * 16 + l16;
  #pragma unroll
  for (int r = 0; r < 8; ++r) {
    const int m = r + 8 * half;
    S[((size_t)tileM * 16 + m) * N_NODES + nCol] = acc[r];
  }
}

// ---------------- fused structural combine + top-32 per row ----------------
__global__ void __launch_bounds__(256)
k_combine_topk(const float* __restrict__ S, const float* __restrict__ adj,
               float* __restrict__ out, int* __restrict__ flags) {
  __shared__ float s[N_NODES];       // 16 KB: full score row
  __shared__ int   nbr[1024];
  __shared__ int   nnbr;
  __shared__ float rmax[256];
  __shared__ int   ridx[256];

  const int row = blockIdx.x;
  const int t   = threadIdx.x;
  if (t == 0) nnbr = 0;

  // s = 1.5 * S_sem[row,:]   (1.0 semantic + 0.5 from the I in A_hat)
  const float* Srow = S + (size_t)row * N_NODES;
  for (int j = t; j < N_NODES; j += 256) s[j] = 1.5f * Srow[j];
  __syncthreads();

  // collect sparse neighbor list from adj[row,:]
  const float* Arow = adj + (size_t)row * N_NODES;
  for (int j = t; j < N_NODES; j += 256) {
    if (Arow[j] != 0.0f) {
      int p = atomicAdd(&nnbr, 1);
      if (p < 1024) nbr[p] = j;
    }
  }
  __syncthreads();

  // s += 0.5 * S_sem[j,:] for each neighbor (L2-hot rows; disjoint columns
  // per thread, so no sync needed inside the loop)
  const int cnt = (nnbr > 1024) ? 1024 : nnbr;
  for (int q = 0; q < cnt; ++q) {
    const float* Sj = S + (size_t)nbr[q] * N_NODES;
    for (int j = t; j < N_NODES; j += 256) s[j] += 0.5f * Sj[j];
  }
  __syncthreads();

  if (t == 0) s[row] = -__builtin_inff();
  __syncthreads();

  // iterative top-32: block-wide argmax, extract, suppress
  for (int kk = 0; kk < TOPK; ++kk) {
    float m = -__builtin_inff();
    int   mi = N_NODES;
    for (int j = t; j < N_NODES; j += 256) {
      float v = s[j];
      if (v > m || (v == m && j < mi)) { m = v; mi = j; }
    }
    rmax[t] = m; ridx[t] = mi;
    __syncthreads();
    #pragma unroll
    for (int off = 128; off > 0; off >>= 1) {
      if (t < off) {
        float mo = rmax[t + off]; int io = ridx[t + off];
        if (mo > rmax[t] || (mo == rmax[t] && io < ridx[t])) {
          rmax[t] = mo; ridx[t] = io;
        }
      }
      __syncthreads();
    }
    if (t == 0) {
      const int bi = ridx[0];
      out[OUT_VAL_OFF + (size_t)row * TOPK + kk] = rmax[0];
      out[OUT_IDX_OFF + (size_t)row * TOPK + kk] = (float)row;          // row idx
      out[OUT_IDX_OFF + (size_t)N_NODES * TOPK
                      + (size_t)row * TOPK + kk] = (float)bi;           // col idx
      flags[bi] = 1;
      s[bi] = -__builtin_inff();
    }
    __syncthreads();
  }
}

// ---------------- sorted unique of selected node ids -----------------------
__global__ void __launch_bounds__(256)
k_unique(const int* __restrict__ flags, float* __restrict__ out) {
  __shared__ int scan[256];
  __shared__ int total;
  const int t = threadIdx.x;
  int c = 0;
  #pragma unroll
  for (int k = 0; k < 16; ++k) c += flags[t * 16 + k];
  scan[t] = c;
  __syncthreads();
  // inclusive Hillis-Steele scan over 256 per-thread counts
  for (int off = 1; off < 256; off <<= 1) {
    int v   = scan[t];
    int add = (t >= off) ? scan[t - off] : 0;
    __syncthreads();
    scan[t] = v + add;
    __syncthreads();
  }
  const int excl = scan[t] - c;
  if (t == 255) total = scan[255];
  __syncthreads();
  int pos = excl;
  #pragma unroll
  for (int k = 0; k < 16; ++k) {
    const int id = t * 16 + k;
    if (flags[id]) out[OUT_SAMP_OFF + pos++] = (float)id;
  }
  for (int p = total + t; p < N_NODES; p += 256)
    out[OUT_SAMP_OFF + p] = (float)N_NODES;  // jnp.unique fill_value = n
}

// ---------------------------------------------------------------------------
extern "C" void kernel_launch(void* const* d_in, const int* in_sizes, int n_in,
                              void* d_out, int out_size, void* d_ws,
                              size_t ws_size, hipStream_t stream) {
  (void)in_sizes; (void)n_in; (void)out_size; (void)ws_size;
  const float* embeds = (const float*)d_in[0];
  const float* adj    = (const float*)d_in[1];
  float* out   = (float*)d_out;
  float* ws    = (float*)d_ws;
  float* X     = ws + WS_X_OFF;
  float* S     = ws + WS_S_OFF;
  int*   flags = (int*)(ws + WS_FLAG_OFF);

  k_normalize<<<N_NODES, 128, 0, stream>>>(embeds, X, flags);

  dim3 g1(32, 256);  // 32 tileN-groups (8 waves each) x 256 tileM
  k_gemm_xxt<<<g1, 256, 0, stream>>>(X, S);

  k_combine_topk<<<N_NODES, 256, 0, stream>>>(S, adj, out, flags);

  k_unique<<<1, 256, 0, stream>>>(flags, out);
}